// AdaptiveModulatedConv3d_20641612824992
// MI455X (gfx1250) — compile-verified
//
#include <hip/hip_runtime.h>

#define BS 8
#define C_IN 64
#define C_OUT 64
#define K3 27
#define KW (C_IN * K3)        // 1728 = GEMM K dimension
#define STYLE_DIM 512
#define BANKN 4
#define DD 32
#define HH 32
#define WWD 32
#define SPATIAL (DD * HH * WWD)   // 32768
#define EPSV 1e-8f

#define KT 32                 // K chunk staged in LDS
#define NTILE 128             // spatial positions per block (N)
#define MTILE 64              // out channels per block (M)

typedef __attribute__((ext_vector_type(2))) float v2f;
typedef __attribute__((ext_vector_type(8))) float v8f;

// workspace layout in floats
#define WS_FWT 0              // [BS][BANKN]
#define WS_MOD 64             // [BS][C_IN]
#define WS_MW  1024           // [BS][C_OUT][KW]

// ---------------- Kernel 1: style math (softmax bank weights + modulation) ----
__global__ __launch_bounds__(256) void style_kernel(
    const float* __restrict__ w, const float* __restrict__ fw,
    const float* __restrict__ fb, const float* __restrict__ mw_,
    const float* __restrict__ mb, float* __restrict__ ws) {
  __shared__ float sw[STYLE_DIM];
  __shared__ float slog[BANKN];
  const int b = blockIdx.x;
  const int t = threadIdx.x;
  for (int k = t; k < STYLE_DIM; k += 256) sw[k] = w[b * STYLE_DIM + k];
  __syncthreads();
  if (t < BANKN) {
    float acc = fb[t];
    for (int k = 0; k < STYLE_DIM; ++k) acc += sw[k] * fw[t * STYLE_DIM + k];
    slog[t] = acc;
  }
  __syncthreads();
  if (t == 0) {
    float m = slog[0];
    for (int n = 1; n < BANKN; ++n) m = fmaxf(m, slog[n]);
    float e[BANKN], s = 0.f;
    for (int n = 0; n < BANKN; ++n) { e[n] = __expf(slog[n] - m); s += e[n]; }
    for (int n = 0; n < BANKN; ++n) ws[WS_FWT + b * BANKN + n] = e[n] / s;
  }
  if (t < C_IN) {
    float acc = mb[t];
    for (int k = 0; k < STYLE_DIM; ++k) acc += sw[k] * mw_[t * STYLE_DIM + k];
    ws[WS_MOD + b * C_IN + t] = acc;
  }
}

// ---------------- Kernel 2: build modulated+demodulated filters --------------
__global__ __launch_bounds__(256) void weight_kernel(
    const float* __restrict__ bank, float* __restrict__ ws) {
  const int o = blockIdx.x;
  const int b = blockIdx.y;
  const int t = threadIdx.x;
  __shared__ float sm[KW];
  __shared__ float red[256];
  const float f0 = ws[WS_FWT + b * BANKN + 0];
  const float f1 = ws[WS_FWT + b * BANKN + 1];
  const float f2 = ws[WS_FWT + b * BANKN + 2];
  const float f3 = ws[WS_FWT + b * BANKN + 3];
  const float* bk = bank + (size_t)o * KW;   // bank[n][o][e], n stride = C_OUT*KW
  float sq = 0.f;
  for (int e = t; e < KW; e += 256) {
    float v = f0 * bk[e] + f1 * bk[(size_t)C_OUT * KW + e] +
              f2 * bk[2 * (size_t)C_OUT * KW + e] +
              f3 * bk[3 * (size_t)C_OUT * KW + e];
    v *= ws[WS_MOD + b * C_IN + e / K3];
    sm[e] = v;
    sq += v * v;
  }
  red[t] = sq;
  __syncthreads();
  for (int s = 128; s > 0; s >>= 1) {
    if (t < s) red[t] += red[t + s];
    __syncthreads();
  }
  const float dem = rsqrtf(red[0] + EPSV);
  float* mwout = ws + WS_MW + (size_t)(b * C_OUT + o) * KW;
  for (int e = t; e < KW; e += 256) mwout[e] = sm[e] * dem;
}

// ---------------- Kernel 3: implicit-GEMM conv via fp32 WMMA -----------------
// Block (256 thr = 8 waves): one sample b, M=64 out-chans x N=128 voxels.
// Waves in 2(M) x 4(N) grid; each wave owns a 32x32 C tile = 4 WMMA accums,
// so each K-step is 4 LDS b64 loads -> 4 v_wmma (1:1 ratio).
__global__ __launch_bounds__(256) void conv_kernel(
    const float* __restrict__ x, const float* __restrict__ ws,
    float* __restrict__ out) {
  __shared__ float As[MTILE][KT + 1];   // [m][k]
  __shared__ float Bs[NTILE][KT + 1];   // [n][k]
  const int b = blockIdx.y;
  const int p0 = blockIdx.x * NTILE;
  const int t = threadIdx.x;
  const int lane = t & 31;
  const int wv = t >> 5;
  const int m_base = (wv & 1) * 32;     // {0,32}
  const int n_base = (wv >> 1) * 32;    // {0,32,64,96}
  const int row = lane & 15;            // M (for A) / N (for B) within 16-tile
  const int kpair = (lane >> 4) * 2;    // lanes>=16 hold K+2,K+3

  const float* A = ws + WS_MW + (size_t)b * C_OUT * KW;
  const float* xb = x + (size_t)b * C_IN * SPATIAL;

  v8f c00 = {}, c01 = {}, c10 = {}, c11 = {};

  for (int k0 = 0; k0 < KW; k0 += KT) {
    // stage A: 64 rows x 32 k, coalesced 32-float rows
    {
      const int m = t >> 5;      // 0..7
      const int kk = t & 31;
#pragma unroll
      for (int i = 0; i < 8; ++i)
        As[m + i * 8][kk] = A[(size_t)(m + i * 8) * KW + k0 + kk];
    }
    // stage B: im2col gather, 32 k x 128 n, consecutive threads -> consecutive n
#pragma unroll
    for (int i = 0; i < 16; ++i) {
      const int idx = i * 256 + t;
      const int pp = idx & 127;
      const int kk = idx >> 7;
      const int k = k0 + kk;
      const int ic = k / K3;
      const int r = k - ic * K3;
      const int kz = r / 9;
      const int r2 = r - kz * 9;
      const int ky = r2 / 3;
      const int kx = r2 - ky * 3;
      const int p = p0 + pp;
      const int dz = (p >> 10) + kz - 1;
      const int dy = ((p >> 5) & 31) + ky - 1;
      const int dx = (p & 31) + kx - 1;
      float v = 0.f;
      if ((unsigned)dz < DD && (unsigned)dy < HH && (unsigned)dx < WWD)
        v = xb[(((size_t)ic * DD + dz) * HH + dy) * WWD + dx];
      Bs[pp][kk] = v;
    }
    __syncthreads();
#pragma unroll
    for (int kk = 0; kk < KT; kk += 4) {
      const int kc = kk + kpair;
      v2f a0, a1, b0, b1;
      a0.x = As[m_base + row][kc];
      a0.y = As[m_base + row][kc + 1];
      a1.x = As[m_base + 16 + row][kc];
      a1.y = As[m_base + 16 + row][kc + 1];
      b0.x = Bs[n_base + row][kc];
      b0.y = Bs[n_base + row][kc + 1];
      b1.x = Bs[n_base + 16 + row][kc];
      b1.y = Bs[n_base + 16 + row][kc + 1];
      c00 = __builtin_amdgcn_wmma_f32_16x16x4_f32(false, a0, false, b0,
                                                  (short)0, c00, false, false);
      c01 = __builtin_amdgcn_wmma_f32_16x16x4_f32(false, a0, false, b1,
                                                  (short)0, c01, false, false);
      c10 = __builtin_amdgcn_wmma_f32_16x16x4_f32(false, a1, false, b0,
                                                  (short)0, c10, false, false);
      c11 = __builtin_amdgcn_wmma_f32_16x16x4_f32(false, a1, false, b1,
                                                  (short)0, c11, false, false);
    }
    __syncthreads();
  }

  // C layout: VGPR r -> M = r (+8 for lanes 16..31), N = lane&15
  const int mo = m_base + ((lane >> 4) << 3);
  const int n = p0 + n_base + (lane & 15);
  float* ob = out + (size_t)b * C_OUT * SPATIAL;
#pragma unroll
  for (int r = 0; r < 8; ++r) {
    ob[(size_t)(mo + r) * SPATIAL + n]           = c00[r];
    ob[(size_t)(mo + r) * SPATIAL + n + 16]      = c01[r];
    ob[(size_t)(mo + 16 + r) * SPATIAL + n]      = c10[r];
    ob[(size_t)(mo + 16 + r) * SPATIAL + n + 16] = c11[r];
  }
}

extern "C" void kernel_launch(void* const* d_in, const int* in_sizes, int n_in,
                              void* d_out, int out_size, void* d_ws,
                              size_t ws_size, hipStream_t stream) {
  const float* x    = (const float*)d_in[0];
  const float* w    = (const float*)d_in[1];
  const float* fw   = (const float*)d_in[2];
  const float* fb   = (const float*)d_in[3];
  const float* mw   = (const float*)d_in[4];
  const float* mb   = (const float*)d_in[5];
  const float* bank = (const float*)d_in[6];
  float* out = (float*)d_out;
  float* ws  = (float*)d_ws;

  style_kernel<<<BS, 256, 0, stream>>>(w, fw, fb, mw, mb, ws);
  dim3 g2(C_OUT, BS);
  weight_kernel<<<g2, 256, 0, stream>>>(bank, ws);
  dim3 g3(SPATIAL / NTILE, BS);
  conv_kernel<<<g3, 256, 0, stream>>>(x, ws, out);
}